// ESSRec_49160195670325
// MI455X (gfx1250) — compile-verified
//
#include <hip/hip_runtime.h>
#include <hip/hip_bf16.h>
#include <math.h>

#define DCOL 64

typedef float v2f __attribute__((ext_vector_type(2)));
typedef float v8f __attribute__((ext_vector_type(8)));
typedef unsigned int v4u __attribute__((ext_vector_type(4)));
typedef int v4i __attribute__((ext_vector_type(4)));
typedef int v8i __attribute__((ext_vector_type(8)));

// ---------------------------------------------------------------- utilities

__global__ void fill_f32(float* __restrict__ p, size_t n, float v) {
  size_t i = (size_t)blockIdx.x * blockDim.x + threadIdx.x;
  size_t stride = (size_t)gridDim.x * blockDim.x;
  for (; i < n; i += stride) p[i] = v;
}

// Pt += p1; Nt += n1; Ot += o1; cur = (p1+n1+o1)/3   (fused single pass)
__global__ void accum_mix(float* __restrict__ Pt, float* __restrict__ Nt,
                          float* __restrict__ Ot, float* __restrict__ cur,
                          const float* __restrict__ p1, const float* __restrict__ n1,
                          const float* __restrict__ o1, size_t n) {
  size_t i = (size_t)blockIdx.x * blockDim.x + threadIdx.x;
  size_t stride = (size_t)gridDim.x * blockDim.x;
  for (; i < n; i += stride) {
    float a = p1[i], b = n1[i], c = o1[i];
    Pt[i] += a;
    Nt[i] += b;
    Ot[i] += c;
    cur[i] = (a + b + c) * (1.0f / 3.0f);
  }
}

// ---------------------------------------------------------- degree handling

__global__ void deg_count(const int* __restrict__ rows, float* __restrict__ deg, int nE) {
  int i = blockIdx.x * blockDim.x + threadIdx.x;
  if (i < nE) atomicAdd(&deg[rows[i]], 1.0f);
}

__global__ void invert_eps(float* __restrict__ d, int n) {
  int i = blockIdx.x * blockDim.x + threadIdx.x;
  if (i < n) d[i] = 1.0f / (d[i] + 1e-7f);
}

// ------------------------------------------------------------------- SpMM
// One wave32 per edge; lane handles 2 columns (float2). Out must be zeroed.
// weight = rowinv[rows[e]]   (matches w = invdeg[row] per edge in reference)

__global__ void spmm_rowinv(const int* __restrict__ rows, const int* __restrict__ cols,
                            const float* __restrict__ rowinv, const float* __restrict__ X,
                            float* __restrict__ Out, int nE) {
  int t = blockIdx.x * blockDim.x + threadIdx.x;
  int e = t >> 5;
  int lane = t & 31;
  if (e >= nE) return;
  int r = rows[e], c = cols[e];
  float w = rowinv[r];
  float2 v = ((const float2*)(X + (size_t)c * DCOL))[lane];
  atomicAdd(&Out[(size_t)r * DCOL + 2 * lane + 0], w * v.x);
  atomicAdd(&Out[(size_t)r * DCOL + 2 * lane + 1], w * v.y);
}

// weight = smul * ew[e]; skips zero-weight edges (wave-uniform branch).
// smul folds the post-SpMM /2, /5 scalings of the reference into the edge pass.
__global__ void spmm_ew(const int* __restrict__ rows, const int* __restrict__ cols,
                        const float* __restrict__ ew, const float* __restrict__ X,
                        float* __restrict__ Out, float smul, int nE) {
  int t = blockIdx.x * blockDim.x + threadIdx.x;
  int e = t >> 5;
  int lane = t & 31;
  if (e >= nE) return;
  float w = ew[e];
  if (w == 0.0f) return;
  w *= smul;
  int r = rows[e], c = cols[e];
  float2 v = ((const float2*)(X + (size_t)c * DCOL))[lane];
  atomicAdd(&Out[(size_t)r * DCOL + 2 * lane + 0], w * v.x);
  atomicAdd(&Out[(size_t)r * DCOL + 2 * lane + 1], w * v.y);
}

// ------------------------------------------------- cosine similarity split

__global__ void row_norms(const float* __restrict__ X, float* __restrict__ nrm, int M) {
  int t = blockIdx.x * blockDim.x + threadIdx.x;
  int r = t >> 5;
  int lane = t & 31;
  if (r >= M) return;
  float2 v = ((const float2*)(X + (size_t)r * DCOL))[lane];
  float s = v.x * v.x + v.y * v.y;
  for (int off = 16; off > 0; off >>= 1) s += __shfl_xor(s, off, 32);
  if (lane == 0) nrm[r] = sqrtf(s);
}

// classify each net edge: 1=P, 2=N, 0=O; accumulate class degrees on net_row
__global__ void edge_class(const int* __restrict__ nr, const int* __restrict__ nc,
                           const int* __restrict__ sgn,
                           const float* __restrict__ R, const float* __restrict__ nrm,
                           float* __restrict__ degP, float* __restrict__ degN,
                           float* __restrict__ degO, unsigned char* __restrict__ cls, int nE) {
  int t = blockIdx.x * blockDim.x + threadIdx.x;
  int e = t >> 5;
  int lane = t & 31;
  if (e >= nE) return;
  int a = nr[e], b = nc[e];
  float2 va = ((const float2*)(R + (size_t)a * DCOL))[lane];
  float2 vb = ((const float2*)(R + (size_t)b * DCOL))[lane];
  float s = va.x * vb.x + va.y * vb.y;
  for (int off = 16; off > 0; off >>= 1) s += __shfl_xor(s, off, 32);
  if (lane == 0) {
    float den = fmaxf(nrm[a] * nrm[b], 1e-8f);
    bool sim = (s / den) > 0.5f;
    int sg = sgn[e];
    int c = (sg == 1 && sim) ? 1 : ((sg == -1 && !sim) ? 2 : 0);
    cls[e] = (unsigned char)c;
    float* dst = (c == 1) ? degP : ((c == 2) ? degN : degO);
    atomicAdd(&dst[a], 1.0f);
  }
}

__global__ void edge_weights(const int* __restrict__ nr, const unsigned char* __restrict__ cls,
                             const float* __restrict__ idP, const float* __restrict__ idN,
                             const float* __restrict__ idO,
                             float* __restrict__ wP, float* __restrict__ wN,
                             float* __restrict__ wO, int nE) {
  int e = blockIdx.x * blockDim.x + threadIdx.x;
  if (e >= nE) return;
  int r = nr[e];
  int c = cls[e];
  wP[e] = (c == 1) ? idP[r] : 0.0f;
  wN[e] = (c == 2) ? idN[r] : 0.0f;
  wO[e] = (c == 0) ? idO[r] : 0.0f;
}

// ------------------------------------------------------------- WMMA GEMM
// Out[M x 64] = s1*X1[M x 64] @ W[0:64,:] + s2*X2[M x 64] @ W[64:128,:] + bias
// FP32 WMMA 16x16x4. Block = 128 threads (4 waves); wave w owns 16-col tile.
// W (128x64 f32 = 32 KB) is staged once per block into LDS by the Tensor Data
// Mover (TENSOR_LOAD_TO_LDS, 2D descriptor), synchronized via s_wait_tensorcnt.
// A frag (16x4 f32): lanes 0-15 hold K,K+1; lanes 16-31 hold K+2,K+3 (row = lane%16)
// B frag (4x16 f32): same K split; col = lane%16 -> read from LDS
// C/D (16x16 f32, 8 VGPR): vgpr v -> row v + 8*(lane/16), col = lane%16

__device__ __forceinline__ v8f gemm_khalf(const float* __restrict__ X, float sc,
                                          const float* __restrict__ Wlds,
                                          int mrow, int half, int n, v8f c) {
  const float* arow = X + (size_t)mrow * DCOL;
#pragma unroll
  for (int k = 0; k < 64; k += 4) {
    int ka = k + 2 * half;
    v2f a, b;
    a[0] = arow[ka + 0] * sc;
    a[1] = arow[ka + 1] * sc;
    b[0] = Wlds[(ka + 0) * DCOL + n];
    b[1] = Wlds[(ka + 1) * DCOL + n];
    c = __builtin_amdgcn_wmma_f32_16x16x4_f32(false, a, false, b, (short)0, c, false, false);
  }
  return c;
}

__global__ void __launch_bounds__(128)
gemm_concat_wmma(const float* __restrict__ X1, float s1,
                 const float* __restrict__ X2, float s2,
                 const float* __restrict__ W,     // [128 x 64] row-major
                 const float* __restrict__ bias,  // [64]
                 float* __restrict__ Out, int M) {
  __shared__ float Wlds[128 * DCOL];   // 32 KB

  // --- TDM: wave 0 DMAs the full weight matrix into LDS ---
  if (threadIdx.x < 32) {
    unsigned lds_base = (unsigned)(size_t)&Wlds[0];        // LDS byte offset
    unsigned long long ga = (unsigned long long)(size_t)W; // 57-bit global address
    v4u g0;
    g0[0] = 1u;                                   // count=1, user-mode descriptor
    g0[1] = lds_base;                             // lds_addr
    g0[2] = (unsigned)(ga & 0xffffffffu);         // global_addr[31:0]
    g0[3] = (unsigned)((ga >> 32) & 0x1ffffffu)   // global_addr[56:32]
            | (2u << 30);                         // type = 2 ("image")
    v8i g1;
    g1[0] = (int)(2u << 16);       // workgroup_mask=0, data_size=2 (4 bytes)
    g1[1] = (int)(64u << 16);      // tensor_dim0[15:0] = 64
    g1[2] = (int)(128u << 16);     // tensor_dim0 hi=0, tensor_dim1[15:0] = 128
    g1[3] = (int)(64u << 16);      // tensor_dim1 hi=0, tile_dim0 = 64
    g1[4] = 128;                   // tile_dim1 = 128, tile_dim2 = 0
    g1[5] = 64;                    // tensor_dim0_stride = 64
    g1[6] = 0;                     // stride0 hi, tensor_dim1_stride lo
    g1[7] = 0;                     // tensor_dim1_stride hi
    v4i z4 = {0, 0, 0, 0};
    v8i z8 = {0, 0, 0, 0, 0, 0, 0, 0};
    // clang-23 / therock-10.0 toolchain: 6-arg form
    // (v4u g0, v8i g1, v4i g2, v4i g3, v8i g4, i32 cpol)
    __builtin_amdgcn_tensor_load_to_lds(g0, g1, z4, z4, z8, 0);
    __builtin_amdgcn_s_wait_tensorcnt(0);
  }
  __syncthreads();

  const int lane = threadIdx.x & 31;
  const int wave = threadIdx.x >> 5;   // 0..3 -> column tile
  const int half = lane >> 4;
  const int l16 = lane & 15;
  const int n = wave * 16 + l16;
  const int mbase = blockIdx.x * 16;   // M is a multiple of 16 -> EXEC all ones
  const int mrow = mbase + l16;

  v8f c;
  float bv = bias[n];
#pragma unroll
  for (int v = 0; v < 8; ++v) c[v] = bv;

  c = gemm_khalf(X1, s1, Wlds, mrow, half, n, c);
  c = gemm_khalf(X2, s2, Wlds + 64 * DCOL, mrow, half, n, c);

#pragma unroll
  for (int v = 0; v < 8; ++v)
    Out[(size_t)(mbase + v + 8 * half) * DCOL + n] = c[v];
}

// ---------------------------------------------------------------- launcher

static inline unsigned cdivu(size_t a, size_t b) { return (unsigned)((a + b - 1) / b); }

extern "C" void kernel_launch(void* const* d_in, const int* in_sizes, int n_in,
                              void* d_out, int out_size, void* d_ws, size_t ws_size,
                              hipStream_t stream) {
  (void)n_in; (void)out_size; (void)ws_size;
  const float* user_emb = (const float*)d_in[0];
  const float* item_emb = (const float*)d_in[1];
  const float* W_uc = (const float*)d_in[2];
  const float* b_uc = (const float*)d_in[3];
  const float* W_ic = (const float*)d_in[4];
  const float* b_ic = (const float*)d_in[5];
  const float* W_ui = (const float*)d_in[6];
  const float* b_ui = (const float*)d_in[7];
  const float* W_pn = (const float*)d_in[8];
  const float* b_pn = (const float*)d_in[9];
  const float* W_u  = (const float*)d_in[10];
  const float* b_u  = (const float*)d_in[11];
  const int* inter_row = (const int*)d_in[12];
  const int* inter_col = (const int*)d_in[13];
  const int* net_row   = (const int*)d_in[14];
  const int* net_col   = (const int*)d_in[15];
  const int* net_sign  = (const int*)d_in[16];

  const int U  = in_sizes[0] / DCOL;   // 100000
  const int I  = in_sizes[1] / DCOL;   // 50000
  const int EU = in_sizes[12];         // 1,000,000
  const int EN = in_sizes[14];         // 1,100,000

  // ---- carve workspace (256B aligned chunks) ----
  char* wsp = (char*)d_ws;
  auto carve = [&](size_t bytes) -> char* {
    char* p = wsp;
    wsp += (bytes + 255) & ~(size_t)255;
    return p;
  };
  float* invdegU = (float*)carve((size_t)U * 4);
  float* invdegI = (float*)carve((size_t)I * 4);
  float* degP = (float*)carve((size_t)U * 4);
  float* degN = (float*)carve((size_t)U * 4);
  float* degO = (float*)carve((size_t)U * 4);
  float* rnorm = (float*)carve((size_t)U * 4);
  float* wP = (float*)carve((size_t)EN * 4);
  float* wN = (float*)carve((size_t)EN * 4);
  float* wO = (float*)carve((size_t)EN * 4);
  unsigned char* cls = (unsigned char*)carve((size_t)EN);
  float* bufU[10];
  for (int i = 0; i < 10; ++i) bufU[i] = (float*)carve((size_t)U * DCOL * 4);
  float* bufI0 = (float*)carve((size_t)I * DCOL * 4);
  float* bufI1 = (float*)carve((size_t)I * DCOL * 4);

  float* user_final = (float*)d_out;
  float* item_init  = (float*)d_out + (size_t)U * DCOL;

  const size_t nU = (size_t)U * DCOL;
  const size_t nI = (size_t)I * DCOL;
  const int TB = 256;

  auto FILL = [&](float* p, size_t n) {
    unsigned g = cdivu(n, (size_t)TB * 4); if (g > 16384u) g = 16384u; if (g == 0) g = 1;
    fill_f32<<<g, TB, 0, stream>>>(p, n, 0.0f);
  };
  auto SPMM_INV = [&](const int* r, const int* c, const float* rowinv,
                      const float* X, float* O, int nE) {
    spmm_rowinv<<<cdivu((size_t)nE * 32, TB), TB, 0, stream>>>(r, c, rowinv, X, O, nE);
  };
  auto SPMM_EW = [&](const float* ew, const float* X, float* O, float smul) {
    spmm_ew<<<cdivu((size_t)EN * 32, TB), TB, 0, stream>>>(net_row, net_col, ew, X, O,
                                                           smul, EN);
  };
  auto GEMM = [&](const float* X1, float s1, const float* X2, float s2,
                  const float* W, const float* b, float* O, int M) {
    gemm_concat_wmma<<<M / 16, 128, 0, stream>>>(X1, s1, X2, s2, W, b, O, M);
  };

  // ---- degrees ----
  FILL(invdegU, U); FILL(invdegI, I);
  deg_count<<<cdivu(EU, TB), TB, 0, stream>>>(inter_row, invdegU, EU);
  deg_count<<<cdivu(EU, TB), TB, 0, stream>>>(inter_col, invdegI, EU);
  invert_eps<<<cdivu(U, TB), TB, 0, stream>>>(invdegU, U);
  invert_eps<<<cdivu(I, TB), TB, 0, stream>>>(invdegI, I);

  // ---- user-item graph convolution ----
  FILL(bufU[0], nU);                                           // item_rep0 = acm(item_emb)
  SPMM_INV(inter_row, inter_col, invdegU, item_emb, bufU[0], EU);
  FILL(bufI0, nI);                                             // user_rep = acmT(user_emb)
  SPMM_INV(inter_col, inter_row, invdegI, user_emb, bufI0, EU);
  FILL(bufI1, nI);                                             // acmT(item_rep0)
  SPMM_INV(inter_col, inter_row, invdegI, bufU[0], bufI1, EU);
  GEMM(item_emb, 1.0f, bufI1, 1.0f, W_ic, b_ic, item_init, I); // item_init -> d_out
  FILL(bufU[1], nU);                                           // acm(user_rep)
  SPMM_INV(inter_row, inter_col, invdegU, bufI0, bufU[1], EU);
  GEMM(user_emb, 1.0f, bufU[1], 1.0f, W_uc, b_uc, bufU[2], U); // user_init
  FILL(bufU[3], nU);                                           // item_rep = acm(item_init)
  SPMM_INV(inter_row, inter_col, invdegU, item_init, bufU[3], EU);
  GEMM(bufU[2], 1.0f, bufU[3], 1.0f, W_ui, b_ui, bufU[4], U);  // E

  // ---- P/N/O edge split on item_rep (bufU[3]) ----
  row_norms<<<cdivu((size_t)U * 32, TB), TB, 0, stream>>>(bufU[3], rnorm, U);
  FILL(degP, U); FILL(degN, U); FILL(degO, U);
  edge_class<<<cdivu((size_t)EN * 32, TB), TB, 0, stream>>>(
      net_row, net_col, net_sign, bufU[3], rnorm, degP, degN, degO, cls, EN);
  invert_eps<<<cdivu(U, TB), TB, 0, stream>>>(degP, U);
  invert_eps<<<cdivu(U, TB), TB, 0, stream>>>(degN, U);
  invert_eps<<<cdivu(U, TB), TB, 0, stream>>>(degO, U);
  edge_weights<<<cdivu(EN, TB), TB, 0, stream>>>(net_row, cls, degP, degN, degO,
                                                 wP, wN, wO, EN);

  // ---- signed GNN: GNN_K=2 outer, GNN_LAYERS=2 inner ----
  // layer-2 /2 and /5 scalings folded into the SpMM edge pass (smul);
  // Pt/Nt/Ot accumulation and cur=(P+N+O)/3 fused into one accum_mix pass.
  float* Pt = bufU[7]; float* Nt = bufU[8]; float* Ot = bufU[9];
  float* cur = bufU[4];  // E
  FILL(Pt, nU); FILL(Nt, nU); FILL(Ot, nU);
  for (int k = 0; k < 2; ++k) {
    float* p  = bufU[0]; float* n  = bufU[1]; float* o  = bufU[2];
    float* p1 = bufU[3]; float* n1 = bufU[5]; float* o1 = bufU[6];
    FILL(p, nU);  SPMM_EW(wP, cur, p, 1.0f);
    FILL(n, nU);  SPMM_EW(wN, cur, n, 1.0f);
    FILL(o, nU);  SPMM_EW(wO, cur, o, 1.0f);
    FILL(p1, nU); SPMM_EW(wP, p, p1, 0.5f); SPMM_EW(wN, n, p1, 0.5f);
    FILL(n1, nU); SPMM_EW(wN, p, n1, 0.5f); SPMM_EW(wP, n, n1, 0.5f);
    FILL(o1, nU); SPMM_EW(wP, o, o1, 0.2f); SPMM_EW(wN, o, o1, 0.2f);
                  SPMM_EW(wO, p, o1, 0.2f); SPMM_EW(wO, n, o1, 0.2f);
                  SPMM_EW(wO, o, o1, 0.2f);
    {
      unsigned g = cdivu(nU, (size_t)TB * 4); if (g > 16384u) g = 16384u;
      accum_mix<<<g, TB, 0, stream>>>(Pt, Nt, Ot, cur, p1, n1, o1, nU);
    }
  }

  // ---- readout; /GNN_K folded into GEMM input scales ----
  GEMM(Pt, 0.5f, Nt, 0.5f, W_pn, b_pn, bufU[0], U);            // user_pn
  GEMM(bufU[0], 1.0f, Ot, 0.5f, W_u, b_u, user_final, U);      // user_final -> d_out
}